// EqFrePBC_37890201485374
// MI455X (gfx1250) — compile-verified
//
#include <hip/hip_runtime.h>
#include <hip/hip_bf16.h>

typedef __attribute__((ext_vector_type(2))) float v2f;
typedef __attribute__((ext_vector_type(8))) float v8f;

#define L_FFT 2048
#define L_MASK 2047
#define NMODE 2
#define KMAX 1600
#define CROP 20
#define L_OUT (L_FFT - 2 * CROP)

// ---------------------------------------------------------------------------
// CDNA5 async memory->LDS copy (ASYNCcnt-tracked, bypasses VGPRs).
// LDS destination offset = low 32 bits of the generic shared pointer
// (see ISA 10.2).
// ---------------------------------------------------------------------------
__device__ inline void async_load_lds_b128(void* lds_dst, const void* gsrc) {
    unsigned lds_off = (unsigned)(unsigned long long)lds_dst;
    asm volatile("global_load_async_to_lds_b128 %0, %1, off"
                 :: "v"(lds_off), "v"(gsrc) : "memory");
}
__device__ inline void wait_asynccnt0() {
    asm volatile("s_wait_asynccnt 0x0" ::: "memory");
}

// ---------------------------------------------------------------------------
// Build the (n1,n2) index pairs on device (constant derived in Python, not an
// input). M=41 -> n1,n2 in [-20,20); |n1*n2| <= RHO*(M//2) = 410.
// ---------------------------------------------------------------------------
__global__ void build_pairs_kernel(int2* __restrict__ pairs, int K) {
    if (threadIdx.x == 0 && blockIdx.x == 0) {
        int idx = 0;
        for (int n1 = -20; n1 < 20; ++n1) {
            for (int n2 = -20; n2 < 20; ++n2) {
                int prod = n1 * n2;
                if (prod < 0) prod = -prod;
                if (prod <= 410) {
                    if (idx < K) pairs[idx] = make_int2(n1, n2);
                    ++idx;
                }
            }
        }
    }
}

// ---------------------------------------------------------------------------
// Radix-2 Stockham FFT over an LDS ping-pong pair. 11 stages for L=2048,
// 1024 butterflies/stage over 256 threads. Returns pointer holding result.
// sign = -1 forward, +1 inverse (no scaling here).
// ---------------------------------------------------------------------------
__device__ inline float2* fft_lds(float2* X, float2* Y, int tid, float sign) {
    int s = 1, t = 0;
    for (int ncur = L_FFT; ncur >= 2; ncur >>= 1) {
        int m = ncur >> 1;
        float ang_step = sign * 6.28318530717958647692f / (float)ncur;
        for (int i = tid; i < (L_FFT / 2); i += 256) {
            int p = i >> t;
            int q = i & (s - 1);
            float2 a = X[q + s * p];
            float2 c = X[q + s * (p + m)];
            float th = ang_step * (float)p;
            float sn, cs;
            __sincosf(th, &sn, &cs);
            float2 su = {a.x + c.x, a.y + c.y};
            float2 d  = {a.x - c.x, a.y - c.y};
            Y[q + s * (2 * p)]     = su;
            Y[q + s * (2 * p + 1)] = make_float2(d.x * cs - d.y * sn,
                                                 d.x * sn + d.y * cs);
        }
        __syncthreads();
        float2* tmp = X; X = Y; Y = tmp;
        s <<= 1; ++t;
    }
    return X;
}

// ---------------------------------------------------------------------------
// Forward FFT: one workgroup per (b,n). x layout [B,L,N] fp32.
// Output xf as float2 [B][L][N] (so (re0,im0,re1,im1) per l is 16B aligned).
// ---------------------------------------------------------------------------
__global__ void fft_fwd_kernel(const float* __restrict__ x_real,
                               const float* __restrict__ x_imag,
                               float2* __restrict__ xf) {
    __shared__ float2 bufA[L_FFT];
    __shared__ float2 bufB[L_FFT];
    int b = blockIdx.x >> 1;
    int n = blockIdx.x & 1;
    int tid = threadIdx.x;
    for (int l = tid; l < L_FFT; l += 256) {
        int gi = (b * L_FFT + l) * NMODE + n;
        bufA[l] = make_float2(x_real[gi], x_imag[gi]);
    }
    __syncthreads();
    float2* R = fft_lds(bufA, bufB, tid, -1.0f);
    for (int l = tid; l < L_FFT; l += 256) {
        xf[(b * L_FFT + l) * NMODE + n] = R[l];
    }
}

// ---------------------------------------------------------------------------
// PBC core: delta[b,l,n] = sum_k E[b,l,k,n] * w[k]
//   E = (g1_0*conj(g12_0) + g1_1*conj(g12_1)) * g2_n   (all rolls of xf)
// Complex K-dot expanded into a real GEMM on V_WMMA_F32_16X16X4_F32:
//   A rows = 16 l's, K' = {k.re, k.im} x 2 pairs per instruction
//   B col0 = (w_r, -w_i) -> delta_re ; col1 = (w_i, w_r) -> delta_im
// Per ISA A-layout: lanes 0..15 supply K'=0,1 (pair k0); lanes 16..31 K'=2,3.
// Two independent accumulators (4 pairs / 2 WMMAs per trip) so consecutive
// XDL ops are not RAW-dependent and co-execute with the VALU E-computation.
// Block = 256 threads = 8 waves: wave -> (l-tile of 16, mode n).
// Grid: x = L/64 chunks, y = B.
// ---------------------------------------------------------------------------
__global__ void pbc_wmma_kernel(const float2* __restrict__ xf,
                                const int2* __restrict__ pairs,
                                const float* __restrict__ wR,
                                const float* __restrict__ wI,
                                int K,
                                float* __restrict__ delta) {
    __shared__ float4 sxf[L_FFT];   // (re0,im0,re1,im1) per l : 32 KB
    __shared__ float2 sw[KMAX];     // (w_r, w_i)              : 12.8 KB
    __shared__ int2   sp[KMAX];     // (n1, n2)                : 12.8 KB

    int tid = threadIdx.x;
    int b = blockIdx.y;
    int Kc = K < KMAX ? K : KMAX;

    // Async DMA the spectrum tile straight into LDS (no VGPR round-trip).
    const float4* xf4 = (const float4*)(xf + (size_t)b * L_FFT * NMODE);
    for (int l = tid; l < L_FFT; l += 256) {
        async_load_lds_b128(&sxf[l], &xf4[l]);
    }
    for (int i = tid; i < Kc; i += 256) {
        sw[i] = make_float2(wR[i], wI[i]);
        sp[i] = pairs[i];
    }
    wait_asynccnt0();
    __syncthreads();

    int lane = tid & 31;
    int wv   = tid >> 5;
    int n    = wv & 1;          // polarization mode handled by this wave
    int tile = wv >> 1;         // which 16-row l tile in the 64-l block chunk
    int half = lane >> 4;       // 0: pair k0 (K'=0,1)  1: pair k1 (K'=2,3)
    int col  = lane & 15;       // B/D column owned by this lane
    int lrow = blockIdx.x * 64 + tile * 16 + (lane & 15); // A-matrix row l
    bool c0 = (col == 0), c1 = (col == 1);

    v8f acc0 = {};
    v8f acc1 = {};
    for (int kb = 0; kb < Kc; kb += 4) {
        #pragma unroll
        for (int u = 0; u < 2; ++u) {
            int p  = kb + 2 * u + half;
            int pc = p < Kc ? p : 0;
            int2 pr = sp[pc];
            int i1  = (lrow - pr.x) & L_MASK;
            int i12 = (lrow - pr.x - pr.y) & L_MASK;
            int i2  = (lrow - pr.y) & L_MASK;
            float4 g1  = sxf[i1];
            float4 g12 = sxf[i12];
            float4 g2  = sxf[i2];
            // S = g1_0*conj(g12_0) + g1_1*conj(g12_1)
            float Sr = g1.x * g12.x + g1.y * g12.y + g1.z * g12.z + g1.w * g12.w;
            float Si = g1.y * g12.x - g1.x * g12.y + g1.w * g12.z - g1.z * g12.w;
            float g2r = n ? g2.z : g2.x;
            float g2i = n ? g2.w : g2.y;
            v2f A;
            A[0] = Sr * g2r - Si * g2i;   // E_re  (K' even slot)
            A[1] = Sr * g2i + Si * g2r;   // E_im  (K' odd slot)
            float2 w2 = sw[pc];
            float wr = (p < Kc) ? w2.x : 0.0f;
            float wi = (p < Kc) ? w2.y : 0.0f;
            v2f B;
            B[0] = c0 ? wr  : (c1 ? wi : 0.0f);  // re-slot row
            B[1] = c0 ? -wi : (c1 ? wr : 0.0f);  // im-slot row
            if (u == 0) {
                acc0 = __builtin_amdgcn_wmma_f32_16x16x4_f32(
                    false, A, false, B, (short)0, acc0, false, false);
            } else {
                acc1 = __builtin_amdgcn_wmma_f32_16x16x4_f32(
                    false, A, false, B, (short)0, acc1, false, false);
            }
        }
    }

    // D layout: VGPR v, lanes 0-15 -> row v, lanes 16-31 -> row v+8.
    // Column 0 = delta_re, column 1 = delta_im.
    if (col < 2) {
        #pragma unroll
        for (int v = 0; v < 8; ++v) {
            int row = v + half * 8;
            int li  = blockIdx.x * 64 + tile * 16 + row;
            delta[(((size_t)b * L_FFT + li) * NMODE + n) * 2 + col] =
                acc0[v] + acc1[v];
        }
    }
}

// ---------------------------------------------------------------------------
// Inverse FFT of delta (+bias in freq domain), scale by P[b], add x0, crop.
// out layout [B, L-40, N, 2].
// ---------------------------------------------------------------------------
__global__ void ifft_finalize_kernel(const float* __restrict__ delta,
                                     const float* __restrict__ x_real,
                                     const float* __restrict__ x_imag,
                                     const float* __restrict__ task_info,
                                     const float* __restrict__ fc_br,
                                     const float* __restrict__ fc_bi,
                                     float* __restrict__ out) {
    __shared__ float2 bufA[L_FFT];
    __shared__ float2 bufB[L_FFT];
    int b = blockIdx.x >> 1;
    int n = blockIdx.x & 1;
    int tid = threadIdx.x;
    float br = fc_br[0], bi = fc_bi[0];
    for (int l = tid; l < L_FFT; l += 256) {
        size_t di = (((size_t)b * L_FFT + l) * NMODE + n) * 2;
        bufA[l] = make_float2(delta[di] + br, delta[di + 1] + bi);
    }
    __syncthreads();
    float2* R = fft_lds(bufA, bufB, tid, +1.0f);

    // P = 1e-3 * 10^(dBm/10) / N
    float P = 1e-3f * __powf(10.0f, task_info[b * 4] * 0.1f) * (1.0f / NMODE);
    const float inv_l = 1.0f / (float)L_FFT;
    for (int t = CROP + tid; t < L_FFT - CROP; t += 256) {
        float dr = R[t].x * inv_l;
        float di = R[t].y * inv_l;
        int gi = (b * L_FFT + t) * NMODE + n;
        float xr = x_real[gi];
        float xi = x_imag[gi];
        size_t o = (((size_t)b * L_OUT + (t - CROP)) * NMODE + n) * 2;
        out[o]     = xr + dr * P;
        out[o + 1] = xi + di * P;
    }
}

// ---------------------------------------------------------------------------
extern "C" void kernel_launch(void* const* d_in, const int* in_sizes, int n_in,
                              void* d_out, int out_size, void* d_ws, size_t ws_size,
                              hipStream_t stream) {
    const float* x_real    = (const float*)d_in[0];
    const float* x_imag    = (const float*)d_in[1];
    const float* task_info = (const float*)d_in[2];
    const float* fc_wr     = (const float*)d_in[3];
    const float* fc_wi     = (const float*)d_in[4];
    const float* fc_br     = (const float*)d_in[5];
    const float* fc_bi     = (const float*)d_in[6];
    int K = in_sizes[3];   // 1600 for defaults

    // Workspace layout:
    //   xf    : B*L*N float2  = 64 KB
    //   delta : B*L*N*2 float = 64 KB
    //   pairs : K int2        = 12.8 KB
    char* ws = (char*)d_ws;
    float2* xf   = (float2*)ws;
    float* delta = (float*)(ws + 64 * 1024);
    int2* pairs  = (int2*)(ws + 128 * 1024);

    build_pairs_kernel<<<1, 1, 0, stream>>>(pairs, K);
    fft_fwd_kernel<<<2 * NMODE, 256, 0, stream>>>(x_real, x_imag, xf);
    pbc_wmma_kernel<<<dim3(L_FFT / 64, 2), 256, 0, stream>>>(
        xf, pairs, fc_wr, fc_wi, K, delta);
    ifft_finalize_kernel<<<2 * NMODE, 256, 0, stream>>>(
        delta, x_real, x_imag, task_info, fc_br, fc_bi, (float*)d_out);
}